// ArchitectureGNN_4844723110065
// MI455X (gfx1250) — compile-verified
//
#include <hip/hip_runtime.h>
#include <math.h>

typedef __attribute__((ext_vector_type(2))) float v2f;
typedef __attribute__((ext_vector_type(8))) float v8f;

#define GAT_D 64

// ---------- monotone float<->uint transform for atomicMax on floats ----------
__device__ __forceinline__ unsigned f2ord(float f) {
    int b = __float_as_int(f);
    return (unsigned)b ^ (unsigned)((b >> 31) | 0x80000000);
}
__device__ __forceinline__ float ord2f(unsigned u) {
    int b = (u & 0x80000000u) ? (int)(u ^ 0x80000000u) : ~(int)u;
    return __int_as_float(b);
}

// ---------- fp32 WMMA GEMM: C[M,N] = A[M,K] * B[K,N] ----------
// One wave computes a 16x64 strip: 4 accumulators share one A fragment per
// k-step, so 4 independent v_wmma issue per A load (hides s_wait_loadcnt).
__global__ void gat_gemm_wmma(const float* __restrict__ A,
                              const float* __restrict__ B,
                              float* __restrict__ C,
                              int M, int K, int N) {
    const int lane  = threadIdx.x;                        // 0..31
    const int grp   = blockIdx.x * blockDim.y + threadIdx.y;  // 64-col group
    const int mtile = blockIdx.y;
    const int n0 = grp * 64;
    const int m0 = mtile * 16;
    if (n0 >= N || m0 >= M) return;                       // wave-uniform

    const int l15 = lane & 15;
    const int kh  = (lane >> 4) << 1;                     // 0 or 2
    const float* Arow = A + (size_t)(m0 + l15) * K;
    v8f acc0 = {0.f,0.f,0.f,0.f,0.f,0.f,0.f,0.f};
    v8f acc1 = acc0, acc2 = acc0, acc3 = acc0;

    for (int k = 0; k < K; k += 4) {
        // A 16x4 fragment: lanes 0-15 hold K=k,k+1 ; lanes 16-31 hold K=k+2,k+3
        v2f a;
        a.x = Arow[k + kh];
        a.y = Arow[k + kh + 1];
        // Four B 4x16 fragments (columns n0+0..63), rows striped across lanes
        const float* Bp = B + (size_t)(k + kh) * N + n0 + l15;
        v2f b0, b1, b2, b3;
        b0.x = Bp[0];  b0.y = Bp[(size_t)N];
        b1.x = Bp[16]; b1.y = Bp[(size_t)N + 16];
        b2.x = Bp[32]; b2.y = Bp[(size_t)N + 32];
        b3.x = Bp[48]; b3.y = Bp[(size_t)N + 48];
        acc0 = __builtin_amdgcn_wmma_f32_16x16x4_f32(false, a, false, b0, (short)0, acc0, false, false);
        acc1 = __builtin_amdgcn_wmma_f32_16x16x4_f32(false, a, false, b1, (short)0, acc1, false, false);
        acc2 = __builtin_amdgcn_wmma_f32_16x16x4_f32(false, a, false, b2, (short)0, acc2, false, false);
        acc3 = __builtin_amdgcn_wmma_f32_16x16x4_f32(false, a, false, b3, (short)0, acc3, false, false);
    }

    // C/D 16x16 f32: VGPR v, lane l -> row m0 + v + 8*(l>>4), col tile_n0 + (l&15)
    float* Cp = C + (size_t)(m0 + ((lane >> 4) << 3)) * N + n0 + l15;
#pragma unroll
    for (int v = 0; v < 8; ++v) {
        Cp[(size_t)v * N + 0]  = acc0[v];
        Cp[(size_t)v * N + 16] = acc1[v];
        Cp[(size_t)v * N + 32] = acc2[v];
        Cp[(size_t)v * N + 48] = acc3[v];
    }
}

// ---------- per-node attention coefficients: es/ed [N,H] ----------
__global__ void gat_attn_coef(const float* __restrict__ h,
                              const float* __restrict__ a_s,
                              const float* __restrict__ a_d,
                              float* __restrict__ es, float* __restrict__ ed,
                              int N, int Hh) {
    int t = blockIdx.x * blockDim.x + threadIdx.x;
    if (t >= N * Hh) return;
    int n  = t / Hh;
    int hh = t - n * Hh;
    const float* hp = h   + ((size_t)n * Hh + hh) * GAT_D;
    const float* as = a_s + (size_t)hh * GAT_D;
    const float* ad = a_d + (size_t)hh * GAT_D;
    float ss = 0.f, dd = 0.f;
#pragma unroll 8
    for (int d = 0; d < GAT_D; ++d) { float v = hp[d]; ss += v * as[d]; dd += v * ad[d]; }
    es[t] = ss; ed[t] = dd;
}

__device__ __forceinline__ void edge_endpoints(const int* __restrict__ ei, int E,
                                               int e, int& src, int& dst) {
    if (e < E) { src = ei[e]; dst = ei[E + e]; }
    else       { src = e - E; dst = e - E;    }       // self-loops
}

// ---------- pass 1: segment max over dst ----------
__global__ void gat_edge_max(const int* __restrict__ ei, int E, int N,
                             const float* __restrict__ es, const float* __restrict__ ed,
                             unsigned* __restrict__ mT, int Hh) {
    int t = blockIdx.x * blockDim.x + threadIdx.x;
    int total = (E + N) * Hh;
    if (t >= total) return;
    int e = t / Hh, hh = t - e * Hh;
    int src, dst; edge_endpoints(ei, E, e, src, dst);
    float s  = es[src * Hh + hh] + ed[dst * Hh + hh];
    float lr = s > 0.f ? s : 0.2f * s;                 // leaky_relu(0.2)
    atomicMax(&mT[dst * Hh + hh], f2ord(lr));
}

// ---------- pass 2: exp(e - m), segment sum ----------
__global__ void gat_edge_exp(const int* __restrict__ ei, int E, int N,
                             const float* __restrict__ es, const float* __restrict__ ed,
                             const unsigned* __restrict__ mT,
                             float* __restrict__ ex, float* __restrict__ den, int Hh) {
    int t = blockIdx.x * blockDim.x + threadIdx.x;
    int total = (E + N) * Hh;
    if (t >= total) return;
    int e = t / Hh, hh = t - e * Hh;
    int src, dst; edge_endpoints(ei, E, e, src, dst);
    float s  = es[src * Hh + hh] + ed[dst * Hh + hh];
    float lr = s > 0.f ? s : 0.2f * s;
    float v  = expf(lr - ord2f(mT[dst * Hh + hh]));
    ex[t] = v;
    atomicAdd(&den[dst * Hh + hh], v);
}

// ---------- pass 3: normalize ex -> alpha in place ----------
__global__ void gat_edge_alpha(const int* __restrict__ ei, int E, int N,
                               float* __restrict__ ex, const float* __restrict__ den, int Hh) {
    int t = blockIdx.x * blockDim.x + threadIdx.x;
    int total = (E + N) * Hh;
    if (t >= total) return;
    int e = t / Hh, hh = t - e * Hh;
    int src, dst; edge_endpoints(ei, E, e, src, dst);
    (void)src;
    ex[t] = ex[t] / den[dst * Hh + hh];
}

// ---------- pass 4: aggregate agg[dst] += h[src] * alpha (channel-major, coalesced) ----------
__global__ void gat_edge_agg(const int* __restrict__ ei, int E, int N,
                             const float* __restrict__ hfeat,
                             const float* __restrict__ alpha,
                             float* __restrict__ agg, int Hh) {
    int HD = Hh * GAT_D;
    int t = blockIdx.x * blockDim.x + threadIdx.x;     // up to ~218M, fits int32
    int total = (E + N) * HD;
    if (t >= total) return;
    int e = t / HD;
    int c = t - e * HD;
    int hh = c >> 6;                                   // D = 64
    int src, dst; edge_endpoints(ei, E, e, src, dst);
    float w = alpha[e * Hh + hh];
    atomicAdd(&agg[(size_t)dst * HD + c], hfeat[(size_t)src * HD + c] * w);
}

// ---------- epilogue: out = (elu?)(agg + b) ----------
__global__ void gat_epilogue(const float* __restrict__ agg, const float* __restrict__ b,
                             float* __restrict__ out, int N, int HD, int do_elu) {
    int t = blockIdx.x * blockDim.x + threadIdx.x;
    if (t >= N * HD) return;
    int c = t - (t / HD) * HD;
    float v = agg[t] + b[c];
    if (do_elu) v = v > 0.f ? v : (expf(v) - 1.f);
    out[t] = v;
}

// ---------- graph embedding: column mean over N nodes ----------
__global__ void gat_colmean(const float* __restrict__ x, float* __restrict__ out, int N, int C) {
    __shared__ float sm[256];
    int c = blockIdx.x;
    float s = 0.f;
    for (int n = threadIdx.x; n < N; n += blockDim.x) s += x[(size_t)n * C + c];
    sm[threadIdx.x] = s;
    __syncthreads();
    for (int off = 128; off > 0; off >>= 1) {
        if ((int)threadIdx.x < off) sm[threadIdx.x] += sm[threadIdx.x + off];
        __syncthreads();
    }
    if (threadIdx.x == 0) out[c] = sm[0] / (float)N;
}

static inline int cdiv(long long a, long long b) { return (int)((a + b - 1) / b); }

extern "C" void kernel_launch(void* const* d_in, const int* in_sizes, int n_in,
                              void* d_out, int out_size, void* d_ws, size_t ws_size,
                              hipStream_t stream) {
    const float* x   = (const float*)d_in[0];
    const int*   ei  = (const int*)  d_in[1];
    const float* W1  = (const float*)d_in[2];
    const float* a1s = (const float*)d_in[3];
    const float* a1d = (const float*)d_in[4];
    const float* b1  = (const float*)d_in[5];
    const float* W2  = (const float*)d_in[6];
    const float* a2s = (const float*)d_in[7];
    const float* a2d = (const float*)d_in[8];
    const float* b2  = (const float*)d_in[9];
    const float* W3  = (const float*)d_in[10];
    const float* a3s = (const float*)d_in[11];
    const float* a3d = (const float*)d_in[12];
    const float* b3  = (const float*)d_in[13];

    const int N  = in_sizes[0] / 16;   // 50000
    const int E  = in_sizes[1] / 2;    // 800000
    const int Et = E + N;              // with self-loops
    const int H  = 4;

    // ---- workspace carve-out (floats) ----
    float* ws = (float*)d_ws;
    size_t off = 0;
    float*    buf_h    = ws + off; off += (size_t)N * 256;   // x @ W
    float*    buf_agg  = ws + off; off += (size_t)N * 256;   // aggregation target
    float*    buf_feat = ws + off; off += (size_t)N * 256;   // layer output
    float*    es       = ws + off; off += (size_t)N * H;
    float*    ed       = ws + off; off += (size_t)N * H;
    unsigned* mT       = (unsigned*)(ws + off); off += (size_t)N * H;
    float*    den      = ws + off; off += (size_t)N * H;
    float*    alpha    = ws + off; off += (size_t)Et * H;
    (void)ws_size;

    float* node_emb  = (float*)d_out;             // [N,64]
    float* graph_emb = node_emb + (size_t)N * 64; // [64]

    auto run_layer = [&](const float* in_feat, const float* W,
                         const float* as_, const float* ad_, const float* bb,
                         float* out_feat, int Hh, int K, int Ncols, int do_elu) {
        const int HD = Hh * GAT_D;
        // GEMM: buf_h = in_feat @ W   (WMMA f32 16x16x4, 16x64 strip per wave)
        int ngrp = Ncols / 64;                       // 64-col groups (4 or 1)
        int wpb  = ngrp < 4 ? ngrp : 4;              // waves per block
        dim3 gb(32, wpb);
        dim3 gg(cdiv(ngrp, wpb), cdiv(N, 16));
        hipLaunchKernelGGL(gat_gemm_wmma, gg, gb, 0, stream, in_feat, W, buf_h, N, K, Ncols);
        // attention coefficients
        hipLaunchKernelGGL(gat_attn_coef, dim3(cdiv((long long)N * Hh, 256)), dim3(256), 0, stream,
                           buf_h, as_, ad_, es, ed, N, Hh);
        // zero segment state + agg
        hipMemsetAsync(mT,      0, (size_t)N * Hh * sizeof(unsigned), stream);
        hipMemsetAsync(den,     0, (size_t)N * Hh * sizeof(float),    stream);
        hipMemsetAsync(buf_agg, 0, (size_t)N * HD * sizeof(float),    stream);
        // edge passes
        int eth = Et * Hh;
        hipLaunchKernelGGL(gat_edge_max,   dim3(cdiv(eth, 256)), dim3(256), 0, stream,
                           ei, E, N, es, ed, mT, Hh);
        hipLaunchKernelGGL(gat_edge_exp,   dim3(cdiv(eth, 256)), dim3(256), 0, stream,
                           ei, E, N, es, ed, mT, alpha, den, Hh);
        hipLaunchKernelGGL(gat_edge_alpha, dim3(cdiv(eth, 256)), dim3(256), 0, stream,
                           ei, E, N, alpha, den, Hh);
        hipLaunchKernelGGL(gat_edge_agg,   dim3(cdiv((long long)Et * HD, 256)), dim3(256), 0, stream,
                           ei, E, N, buf_h, alpha, buf_agg, Hh);
        // epilogue
        hipLaunchKernelGGL(gat_epilogue, dim3(cdiv((long long)N * HD, 256)), dim3(256), 0, stream,
                           buf_agg, bb, out_feat, N, HD, do_elu);
    };

    // Layer 1: [N,16] -> [N,256], ELU
    run_layer(x,        W1, a1s, a1d, b1, buf_feat, H, 16,  256, 1);
    // Layer 2: [N,256] -> [N,256], ELU
    run_layer(buf_feat, W2, a2s, a2d, b2, buf_feat, H, 256, 256, 1);
    // Layer 3: [N,256] -> [N,64], single head, no ELU -> node_emb
    run_layer(buf_feat, W3, a3s, a3d, b3, node_emb, 1, 256, 64, 0);

    // graph_emb = mean over nodes
    hipLaunchKernelGGL(gat_colmean, dim3(64), dim3(256), 0, stream, node_emb, graph_emb, N, 64);
}